// RegionProposalNetwork_64939905516020
// MI455X (gfx1250) — compile-verified
//
#include <hip/hip_runtime.h>
#include <stdint.h>

// ---------------- static problem configuration ----------------
#define BATCH    8
#define A_TOTAL  242991
#define K_TOTAL  4741
#define POST_N   1000
#define NPAD     8192          // next pow2 >= K_TOTAL for bitonic sort
#define NWORDS   149           // ceil(K_TOTAL/32)
#define NMS_THR  0.7f
#define IMG_W    1216.0f
#define IMG_H    800.0f
#define MIN_SZ   0.001f
#define DWH_CLIP 4.135166556742356f   // log(1000/16)

// level boundaries in anchor space
__device__ __forceinline__ int level_of(int a) {
  if (a < 182400) return 0;
  if (a < 228000) return 1;
  if (a < 239400) return 2;
  if (a < 242250) return 3;
  return 4;
}

// monotone float<->uint key (descending top-k = largest keys)
__device__ __forceinline__ unsigned mono_key(float f) {
  unsigned u = __float_as_uint(f);
  return (u & 0x80000000u) ? ~u : (u | 0x80000000u);
}
__device__ __forceinline__ float unmono_key(unsigned m) {
  unsigned u = (m & 0x80000000u) ? (m ^ 0x80000000u) : ~m;
  return __uint_as_float(u);
}

// ---------------- CDNA5 TDM (async tensor load to LDS) ----------------
typedef unsigned int u32x4 __attribute__((ext_vector_type(4)));
typedef int          i32x4 __attribute__((ext_vector_type(4)));
typedef int          i32x8 __attribute__((ext_vector_type(8)));

// Issue (no wait): 1-row 2D descriptor copying nelem 4-byte elements
// (nelem < 65536) from global memory to LDS, per CDNA5 ISA D#.
// Completion is tracked by TENSORcnt; caller must s_wait_tensorcnt.
__device__ __forceinline__ void tdm_issue_load_f32(unsigned lds_off, const void* gptr,
                                                   unsigned nelem) {
  uint64_t ga = (uint64_t)(uintptr_t)gptr;
  u32x4 g0;
  g0[0] = 1u;                                              // count=1 valid descriptor
  g0[1] = lds_off;                                         // lds_addr
  g0[2] = (unsigned)(ga & 0xFFFFFFFFull);                  // global_addr[31:0]
  g0[3] = (unsigned)((ga >> 32) & 0x1FFFFFFull)            // global_addr[56:32]
        | 0x80000000u;                                     // type=2 ("image")
  i32x8 g1;
  g1[0] = (int)(2u << 16);                                 // data_size=2 (4B), no multicast
  g1[1] = (int)((nelem & 0xFFFFu) << 16);                  // tensor_dim0[15:0]
  g1[2] = (int)((nelem >> 16) | (1u << 16));               // tensor_dim0[31:16] | tensor_dim1=1
  g1[3] = (int)((nelem & 0xFFFFu) << 16);                  // tile_dim0 = nelem
  g1[4] = 1;                                               // tile_dim1=1, tile_dim2=0
  g1[5] = (int)nelem;                                      // tensor_dim0_stride[31:0]
  g1[6] = (int)((nelem & 0xFFFFu) << 16);                  // stride0[47:32]=0 | stride1[15:0]
  g1[7] = 0;
  i32x4 gz = {0, 0, 0, 0};
#if __clang_major__ >= 23
  i32x8 gz8 = {0, 0, 0, 0, 0, 0, 0, 0};
  __builtin_amdgcn_tensor_load_to_lds(g0, g1, gz, gz, gz8, 0);
#else
  __builtin_amdgcn_tensor_load_to_lds(g0, g1, gz, gz, 0);
#endif
}

// ---------------- stage 0: init workspace state ----------------
// state layout (unsigned[200]): [0..39]=prefix, [40..79]=need, [80..119]=tie quota,
// [120..159]=main counter, [160..199]=tie counter.   40 = BATCH*5 (b,l) pairs.
__global__ void rpn_init(unsigned* __restrict__ hist, unsigned* __restrict__ state) {
  int t = blockIdx.x * blockDim.x + threadIdx.x;
  if (t < 40 * 2048) hist[t] = 0u;
  if (t < 200) {
    unsigned v = 0u;
    if (t >= 40 && t < 80) { int l = (t - 40) % 5; v = (l < 4) ? 1000u : 741u; }
    state[t] = v;
  }
}

// ---------------- stage 1: radix-select histogram pass ----------------
__global__ void rpn_hist(const float* __restrict__ obj, unsigned* __restrict__ hist,
                         const unsigned* __restrict__ state,
                         int shift, unsigned mask, unsigned nbins) {
  int t = blockIdx.x * blockDim.x + threadIdx.x;
  if (t >= BATCH * A_TOTAL) return;
  __builtin_prefetch(obj + t + 16384, 0, 0);   // global_prefetch_b8 on gfx1250
  int b = t / A_TOTAL;
  int a = t - b * A_TOTAL;
  unsigned key = mono_key(obj[t]);
  int sl = b * 5 + level_of(a);
  if ((key & mask) == state[sl]) {
    unsigned bin = (key >> shift) & (nbins - 1u);
    atomicAdd(&hist[sl * 2048 + bin], 1u);
  }
}

// ---------------- stage 2: scan histogram, refine threshold ----------------
__global__ void rpn_scan(unsigned* __restrict__ hist, unsigned* __restrict__ state,
                         int shift, int nbins, int last) {
  if (threadIdx.x != 0) return;
  int sl = blockIdx.x;                // 0..39
  unsigned* h = hist + sl * 2048;
  unsigned need = state[40 + sl];
  unsigned cum = 0u;
  int sel = 0;
  for (int bin = nbins - 1; bin >= 0; --bin) {
    unsigned c = h[bin];
    if (cum + c >= need) { sel = bin; break; }
    cum += c;
  }
  state[sl]      |= ((unsigned)sel) << shift;   // extend prefix
  state[40 + sl]  = need - cum;                 // remaining inside selected bin
  if (last) {
    state[80 + sl]  = need - cum;               // tie quota at exact key
    state[120 + sl] = 0u;                       // main counter
    state[160 + sl] = 0u;                       // tie counter
  }
  for (int bin = 0; bin < nbins; ++bin) h[bin] = 0u;   // self-restoring for replay
}

// ---------------- stage 3: compact + fused decode/clip/sigmoid ----------------
__global__ void rpn_compact(const float* __restrict__ obj,
                            const float* __restrict__ deltas,
                            const float* __restrict__ anchors,
                            unsigned* __restrict__ state,
                            float* __restrict__ ws_scores,
                            float* __restrict__ ws_boxes) {
  int t = blockIdx.x * blockDim.x + threadIdx.x;
  if (t >= BATCH * A_TOTAL) return;
  int b = t / A_TOTAL;
  int a = t - b * A_TOTAL;
  float v = obj[t];
  unsigned key = mono_key(v);
  int l = level_of(a);
  int sl = b * 5 + l;
  unsigned T = state[sl];
  bool take = false;
  if (key > T) {
    take = true;
  } else if (key == T) {
    unsigned tk = atomicAdd(&state[160 + sl], 1u);
    if (tk < state[80 + sl]) take = true;
  }
  if (!take) return;
  unsigned pos = atomicAdd(&state[120 + sl], 1u);     // < k by construction
  int slot = l * 1000 + (int)pos;                     // K_PER_LEVEL offsets are l*1000

  const float* d4 = deltas + ((size_t)b * A_TOTAL + a) * 4;
  const float* an = anchors + (size_t)a * 4;
  float a0 = an[0], a1 = an[1], a2 = an[2], a3 = an[3];
  float wa = a2 - a0, ha = a3 - a1;
  float cxa = a0 + 0.5f * wa, cya = a1 + 0.5f * ha;
  float dx = d4[0], dy = d4[1];
  float dw = fminf(d4[2], DWH_CLIP), dh = fminf(d4[3], DWH_CLIP);
  float cx = dx * wa + cxa, cy = dy * ha + cya;
  float w = expf(dw) * wa, h = expf(dh) * ha;
  float x1 = cx - 0.5f * w, y1 = cy - 0.5f * h;
  float x2 = cx + 0.5f * w, y2 = cy + 0.5f * h;
  x1 = fminf(fmaxf(x1, 0.0f), IMG_W);
  x2 = fminf(fmaxf(x2, 0.0f), IMG_W);
  y1 = fminf(fmaxf(y1, 0.0f), IMG_H);
  y2 = fminf(fmaxf(y2, 0.0f), IMG_H);
  float score = 1.0f / (1.0f + expf(-v));
  bool valid = (x2 - x1 >= MIN_SZ) && (y2 - y1 >= MIN_SZ) && (score >= 0.0f);
  int gi = b * K_TOTAL + slot;
  ws_scores[gi] = valid ? score : -__builtin_inff();
  float4 bx = make_float4(x1, y1, x2, y2);
  ((float4*)ws_boxes)[gi] = bx;
}

// ---------------- stage 4: per-image sort + NMS + output ----------------
// LDS carve (dynamic): pairs 64KB | boxes (slot order, TDM target) 75904B |
// keep 608B | scan 1024B
#define SM_PAIR   0
#define SM_BOX    65536
#define SM_KEEP   (SM_BOX + 75904)
#define SM_SCAN   (SM_KEEP + 608)
#define SM_BYTES  (SM_SCAN + 1024 + 64)

extern "C" __global__ __launch_bounds__(1024)
void rpn_sort_nms(const float* __restrict__ ws_scores,
                  const float* __restrict__ ws_boxes,
                  float* __restrict__ out) {
  extern __shared__ __align__(16) unsigned char smem[];
  unsigned long long* s_pair = (unsigned long long*)(smem + SM_PAIR);
  float4*             s_box  = (float4*)(smem + SM_BOX);   // indexed by slot
  unsigned*           s_keep = (unsigned*)(smem + SM_KEEP);
  unsigned*           s_scan = (unsigned*)(smem + SM_SCAN);

  const int b = blockIdx.x;
  const int tid = threadIdx.x;
  const float*  sc_g = ws_scores + (size_t)b * K_TOTAL;
  const float4* bx_g = (const float4*)ws_boxes + (size_t)b * K_TOTAL;

  // --- issue async TDM copy of the 75.9KB box block (slot order) into LDS.
  // The Tensor Data Mover streams it while all waves build + sort the keys.
  if ((tid >> 5) == 0) {        // one wave issues; EXEC ignored by TDM
    tdm_issue_load_f32((unsigned)(uintptr_t)s_box, bx_g, K_TOTAL * 4);
  }

  // --- build (key,slot) pairs from global scores (19KB, L2-resident) ---
  for (int t = tid; t < NPAD; t += 1024) {
    unsigned long long p = 0ull;
    if (t < K_TOTAL) {
      p = ((unsigned long long)mono_key(sc_g[t]) << 32) | (unsigned)t;
    }
    s_pair[t] = p;
  }
  __syncthreads();

  // --- bitonic sort, descending (overlapped with the TDM transfer) ---
  for (unsigned k = 2; k <= NPAD; k <<= 1) {
    for (unsigned j = k >> 1; j > 0; j >>= 1) {
      for (int i = tid; i < NPAD; i += 1024) {
        unsigned p = (unsigned)i ^ j;
        if (p > (unsigned)i) {
          unsigned long long x = s_pair[i], y = s_pair[p];
          bool desc = (((unsigned)i & k) == 0u);
          if (desc ? (x < y) : (x > y)) { s_pair[i] = y; s_pair[p] = x; }
        }
      }
      __syncthreads();
    }
  }

  // --- keep-word init from sorted validity bits (key sign bit) ---
  for (int w = tid; w < 152; w += 1024) {
    unsigned bits = 0u;
    for (int j = 0; j < 32; ++j) {
      int idx = w * 32 + j;
      if (idx < K_TOTAL && (s_pair[idx] >> 63)) bits |= 1u << j;
    }
    s_keep[w] = bits;
  }

  // --- TDM completion: issuing wave drains TENSORcnt, barrier publishes LDS ---
  if ((tid >> 5) == 0) {
    __builtin_amdgcn_s_wait_tensorcnt(0);
  }
  __syncthreads();

  // --- serial-scan NMS, keep bitset in LDS; boxes read via slot indirection ---
  for (int i = 0; i < K_TOTAL; ++i) {
    __syncthreads();
    if ((s_keep[i >> 5] >> (i & 31)) & 1u) {
      unsigned slot_i = (unsigned)(s_pair[i] & 0xFFFFFFFFull);
      unsigned li = slot_i / 1000u;
      float4 bi = s_box[slot_i];
      float areai = (bi.z - bi.x) * (bi.w - bi.y);
      for (int j = i + 1 + tid; j < K_TOTAL; j += 1024) {
        if ((s_keep[j >> 5] >> (j & 31)) & 1u) {
          unsigned slot_j = (unsigned)(s_pair[j] & 0xFFFFFFFFull);
          if (slot_j / 1000u == li) {   // cross-level boxes never suppress (LEVEL_OFFSET trick)
            float4 bj = s_box[slot_j];
            float iw = fmaxf(fminf(bi.z, bj.z) - fmaxf(bi.x, bj.x), 0.0f);
            float ih = fmaxf(fminf(bi.w, bj.w) - fmaxf(bi.y, bj.y), 0.0f);
            float inter = iw * ih;
            float areaj = (bj.z - bj.x) * (bj.w - bj.y);
            float iou = inter / (areai + areaj - inter);
            if (iou > NMS_THR) atomicAnd(&s_keep[j >> 5], ~(1u << (j & 31)));
          }
        }
      }
    }
  }
  __syncthreads();

  // --- inclusive prefix sum of keep-word popcounts ---
  if (tid < 256) s_scan[tid] = (tid < NWORDS) ? (unsigned)__popc(s_keep[tid]) : 0u;
  __syncthreads();
  for (int d = 1; d < 256; d <<= 1) {
    unsigned v = 0u;
    if (tid < 256) { v = s_scan[tid]; if (tid >= d) v += s_scan[tid - d]; }
    __syncthreads();
    if (tid < 256) s_scan[tid] = v;
    __syncthreads();
  }
  unsigned kept_total = s_scan[NWORDS - 1];

  float* out_boxes = out;                        // [8,1000,4]
  float* out_scores = out + BATCH * POST_N * 4;  // [8,1000]

  // --- emit kept proposals in sorted order ---
  for (int w = tid; w < NWORDS; w += 1024) {
    unsigned bits = s_keep[w];
    unsigned pos = (w == 0) ? 0u : s_scan[w - 1];
    while (bits) {
      int bit = __builtin_ctz(bits);
      bits &= bits - 1u;
      if (pos >= POST_N) break;
      int src = w * 32 + bit;
      unsigned long long pr = s_pair[src];
      unsigned slot = (unsigned)(pr & 0xFFFFFFFFull);
      float sc = unmono_key((unsigned)(pr >> 32));
      float4 bx = s_box[slot];
      int o = b * POST_N + (int)pos;
      out_boxes[o * 4 + 0] = bx.x;
      out_boxes[o * 4 + 1] = bx.y;
      out_boxes[o * 4 + 2] = bx.z;
      out_boxes[o * 4 + 3] = bx.w;
      out_scores[o] = sc;
      ++pos;
    }
  }
  // --- zero-pad the tail ---
  int start = (kept_total > POST_N) ? POST_N : (int)kept_total;
  for (int j = start + tid; j < POST_N; j += 1024) {
    int o = b * POST_N + j;
    out_boxes[o * 4 + 0] = 0.0f;
    out_boxes[o * 4 + 1] = 0.0f;
    out_boxes[o * 4 + 2] = 0.0f;
    out_boxes[o * 4 + 3] = 0.0f;
    out_scores[o] = 0.0f;
  }
}

// ---------------- host-side launch ----------------
extern "C" void kernel_launch(void* const* d_in, const int* in_sizes, int n_in,
                              void* d_out, int out_size, void* d_ws, size_t ws_size,
                              hipStream_t stream) {
  (void)in_sizes; (void)n_in; (void)out_size; (void)ws_size;
  const float* obj     = (const float*)d_in[0];   // [8,242991]
  const float* deltas  = (const float*)d_in[1];   // [8,242991,4]
  const float* anchors = (const float*)d_in[2];   // [242991,4]
  float* out = (float*)d_out;                     // boxes [8,1000,4] ++ scores [8,1000]

  char* ws = (char*)d_ws;
  unsigned* hist      = (unsigned*)(ws);                      // 40*2048*4 = 327680
  unsigned* state     = (unsigned*)(ws + 327680);             // 200 u32 (pad 1024)
  float*    ws_scores = (float*)(ws + 327680 + 1024);         // 8*4741*4 = 151712 (pad 151808)
  float*    ws_boxes  = (float*)(ws + 327680 + 1024 + 151808);// 8*4741*16 = 606848

  const int totalT = BATCH * A_TOTAL;
  dim3 blk(256), grd((totalT + 255) / 256);

  rpn_init<<<(40 * 2048 + 255) / 256, 256, 0, stream>>>(hist, state);
  // 3-pass exact radix select on 32-bit monotone keys (11 + 11 + 10 bits)
  rpn_hist<<<grd, blk, 0, stream>>>(obj, hist, state, 21, 0x00000000u, 2048);
  rpn_scan<<<40, 32, 0, stream>>>(hist, state, 21, 2048, 0);
  rpn_hist<<<grd, blk, 0, stream>>>(obj, hist, state, 10, 0xFFE00000u, 2048);
  rpn_scan<<<40, 32, 0, stream>>>(hist, state, 10, 2048, 0);
  rpn_hist<<<grd, blk, 0, stream>>>(obj, hist, state, 0, 0xFFFFFC00u, 1024);
  rpn_scan<<<40, 32, 0, stream>>>(hist, state, 0, 1024, 1);
  // fused compact + decode + clip + sigmoid
  rpn_compact<<<grd, blk, 0, stream>>>(obj, deltas, anchors, state, ws_scores, ws_boxes);
  // one workgroup per image: async TDM box stage || bitonic sort -> NMS -> output
  rpn_sort_nms<<<BATCH, 1024, SM_BYTES, stream>>>(ws_scores, ws_boxes, out);
}